// kNNLoss_32177894981697
// MI455X (gfx1250) — compile-verified
//
#include <hip/hip_runtime.h>

#define B_SZ    32
#define N_PTS   16384
#define NSEED   20
#define KNN     11          // K+1
#define CH      16          // chunks per batch
#define CHUNK   1024        // N_PTS / CH

typedef __attribute__((ext_vector_type(2))) float v2f;
typedef __attribute__((ext_vector_type(8))) float v8f;

__device__ __forceinline__ unsigned rotl32(unsigned x, int r) {
  return (x << r) | (x >> (32 - r));
}

// jax.random.randint(key(42),(32,),0,16384): threefry2x32 with (k0,k1)=(0,42),
// counts iota(32) split into halves; span=2^14 -> bits & 0x3FFF.
__device__ unsigned jax_bits_for_index(int i) {
  unsigned c0, c1; int word;
  if (i < 16) { c0 = (unsigned)i;        c1 = (unsigned)(i + 16); word = 0; }
  else        { c0 = (unsigned)(i - 16); c1 = (unsigned)i;        word = 1; }
  const unsigned k0 = 0u, k1 = 42u;
  const unsigned k2 = 0x1BD11BDAu ^ k0 ^ k1;
  unsigned x0 = c0 + k0, x1 = c1 + k1;
#define TF_R(r) { x0 += x1; x1 = rotl32(x1, (r)); x1 ^= x0; }
  TF_R(13) TF_R(15) TF_R(26) TF_R(6)   x0 += k1; x1 += k2 + 1u;
  TF_R(17) TF_R(29) TF_R(16) TF_R(24)  x0 += k2; x1 += k0 + 2u;
  TF_R(13) TF_R(15) TF_R(26) TF_R(6)   x0 += k0; x1 += k1 + 3u;
  TF_R(17) TF_R(29) TF_R(16) TF_R(24)  x0 += k1; x1 += k2 + 4u;
  TF_R(13) TF_R(15) TF_R(26) TF_R(6)   x0 += k2; x1 += k0 + 5u;
#undef TF_R
  return word == 0 ? x0 : x1;
}

// ---------------------------------------------------------------- FPS
// One block per batch. dist[] lives in dynamic LDS (64 KB). 20 serial
// min-update + block-argmax rounds; emits WMMA B-fragment data per seed:
// (-2sx,-2sy,-2sz,|s|^2), slots 20..31 zeroed (WMMA column padding).
__global__ void fps_kernel(const float* __restrict__ pcs,
                           float* __restrict__ seedsB) {
  extern __shared__ float dist[];           // N_PTS floats
  __shared__ float rv[256];
  __shared__ int   ri[256];
  __shared__ int   s_far;
  __shared__ float s_c[3];
  const int b = blockIdx.x, tid = threadIdx.x;
  const float* P = pcs + (size_t)b * N_PTS * 3;

  for (int i = tid; i < N_PTS; i += 256) dist[i] = 1e10f;
  if (tid == 0) s_far = (int)(jax_bits_for_index(b) & 16383u);
  if (tid >= NSEED && tid < 32) {
    float* o = seedsB + ((size_t)b * 32 + tid) * 4;
    o[0] = 0.f; o[1] = 0.f; o[2] = 0.f; o[3] = 0.f;
  }
  __syncthreads();

  for (int s = 0; s < NSEED; ++s) {
    if (tid == 0) {
      const int f = s_far;
      const float x = P[f*3+0], y = P[f*3+1], z = P[f*3+2];
      s_c[0] = x; s_c[1] = y; s_c[2] = z;
      float* o = seedsB + ((size_t)b * 32 + s) * 4;
      o[0] = -2.f*x; o[1] = -2.f*y; o[2] = -2.f*z; o[3] = x*x + y*y + z*z;
    }
    __syncthreads();
    const float cx = s_c[0], cy = s_c[1], cz = s_c[2];
    float bv = -1.f; int bi = 0x7fffffff;
    for (int i = tid; i < N_PTS; i += 256) {
      const float dx = P[i*3+0]-cx, dy = P[i*3+1]-cy, dz = P[i*3+2]-cz;
      const float d  = dx*dx + dy*dy + dz*dz;
      const float dm = fminf(dist[i], d);
      dist[i] = dm;
      if (dm > bv) { bv = dm; bi = i; }   // first-occurrence argmax semantics
    }
    rv[tid] = bv; ri[tid] = bi;
    __syncthreads();
    for (int off = 128; off > 0; off >>= 1) {
      if (tid < off) {
        const float ov = rv[tid+off]; const int oi = ri[tid+off];
        if (ov > rv[tid] || (ov == rv[tid] && oi < ri[tid])) { rv[tid]=ov; ri[tid]=oi; }
      }
      __syncthreads();
    }
    if (tid == 0) s_far = ri[0];
    __syncthreads();
  }
}

// ---------------------------------------------------- WMMA distance + top-K
// Block = (chunk, batch), 256 thr = 8 waves. Per 16-point tile:
//   A (16x4 f32, 2 VGPR): lanes 0-15 -> (x,y); lanes 16-31 -> (z, |p|^2)
//   B (4x16 f32, 2 VGPR): lanes 0-15 -> (-2sx,-2sy); lanes 16-31 -> (-2sz, 1)
//   D = |p|^2 - 2 p.s ; lane n owns seed column n -> add |s_n|^2 scalar.
// Squared distances staged in LDS; top-11 selection (sqrt deferred: monotone).
__global__ void knn_partial(const float* __restrict__ pcs,
                            const float* __restrict__ seedsB,
                            float* __restrict__ cand) {
  extern __shared__ float smem[];
  float* ldist = smem;                         // CHUNK * NSEED
  float* sel   = smem + CHUNK * NSEED;         // 160 * KNN
  const int b = blockIdx.y, chunk = blockIdx.x, tid = threadIdx.x;
  const int lane = tid & 31, wave = tid >> 5;
  const int n = lane & 15;
  const bool hi = lane >= 16;

  const float* P  = pcs + ((size_t)b * N_PTS + (size_t)chunk * CHUNK) * 3;
  const float* SB = seedsB + (size_t)b * 32 * 4;

  v2f bf0, bf1; float sq0, sq1;
  { const float* s4 = SB + n * 4;
    bf0.x = hi ? s4[2] : s4[0];  bf0.y = hi ? 1.0f : s4[1];  sq0 = s4[3]; }
  { const float* s4 = SB + (16 + n) * 4;     // zeros for seeds >= 20
    bf1.x = hi ? s4[2] : s4[0];  bf1.y = hi ? 1.0f : s4[1];  sq1 = s4[3]; }
  const bool valid1 = (n < (NSEED - 16));

  for (int k = 0; k < 8; ++k) {
    const int tile = wave * 8 + k;
    const float* pp = P + (size_t)(tile * 16 + n) * 3;
    __builtin_prefetch(pp + 16 * 3, 0, 3);     // global_prefetch_b8, next tile
    v2f af;
    if (!hi) { af.x = pp[0]; af.y = pp[1]; }
    else { const float x = pp[0], y = pp[1], z = pp[2];
           af.x = z; af.y = x*x + y*y + z*z; }
    v8f acc0 = {}; v8f acc1 = {};
    acc0 = __builtin_amdgcn_wmma_f32_16x16x4_f32(false, af, false, bf0,
                                                 (short)0, acc0, false, false);
    acc1 = __builtin_amdgcn_wmma_f32_16x16x4_f32(false, af, false, bf1,
                                                 (short)0, acc1, false, false);
    const int pbase = tile * 16 + (hi ? 8 : 0);   // D rows: lanes<16 -> M 0..7
#pragma unroll
    for (int r = 0; r < 8; ++r)
      ldist[(pbase + r) * NSEED + n] = acc0[r] + sq0;
    if (valid1) {
#pragma unroll
      for (int r = 0; r < 8; ++r)
        ldist[(pbase + r) * NSEED + 16 + n] = acc1[r] + sq1;
    }
  }
  __syncthreads();

  // per-(seed, sub-range) top-11 insertion lists: 160 workers
  if (tid < 8 * NSEED) {
    const int j = tid % NSEED, sub = tid / NSEED;
    float list[KNN];
#pragma unroll
    for (int q = 0; q < KNN; ++q) list[q] = 3.0e38f;
    const int p0 = sub * (CHUNK / 8);
    for (int p = p0; p < p0 + CHUNK / 8; ++p) {
      const float d = ldist[p * NSEED + j];
      if (d < list[KNN - 1]) {
        int q = KNN - 1;
        while (q > 0 && list[q - 1] > d) { list[q] = list[q - 1]; --q; }
        list[q] = d;
      }
    }
#pragma unroll
    for (int q = 0; q < KNN; ++q) sel[tid * KNN + q] = list[q];
  }
  __syncthreads();

  // merge 8 sub-lists -> 11 smallest (ascending) per seed for this chunk
  if (tid < NSEED) {
    const int j = tid;
    float* outp = cand + (((size_t)(b * CH + chunk)) * NSEED + j) * KNN;
    for (int it = 0; it < KNN; ++it) {
      float mv = 3.3e38f; int mi = 0;
      for (int sub = 0; sub < 8; ++sub) {
        const int base = (sub * NSEED + j) * KNN;
        for (int q = 0; q < KNN; ++q) {
          const float v = sel[base + q];
          if (v < mv) { mv = v; mi = base + q; }
        }
      }
      sel[mi] = 3.35e38f;                      // mark consumed
      outp[it] = mv;
    }
  }
}

// ------------------------------------------------- merge chunks per (b,seed)
__global__ void knn_merge(const float* __restrict__ cand,
                          float* __restrict__ topd) {
  __shared__ float lin[CH * KNN];              // 176 candidates
  const int blk = blockIdx.x;
  const int b = blk / NSEED, j = blk % NSEED;
  const int tid = threadIdx.x;
  for (int i = tid; i < CH * KNN; i += 64) {
    const int c = i / KNN, q = i % KNN;
    lin[i] = cand[(((size_t)(b * CH + c)) * NSEED + j) * KNN + q];
  }
  __syncthreads();
  if (tid == 0) {
    for (int it = 0; it < KNN; ++it) {
      float mv = 3.3e38f; int mi = 0;
      for (int q = 0; q < CH * KNN; ++q) {
        const float v = lin[q];
        if (v < mv) { mv = v; mi = q; }
      }
      lin[mi] = 3.35e38f;
      topd[((size_t)b * NSEED + j) * KNN + it] = sqrtf(fmaxf(mv, 0.f));
    }
  }
}

// ----------------------------------------------------------- scalar epilogue
__global__ void finalize_kernel(const float* __restrict__ topd,
                                float* __restrict__ out) {
  __shared__ float red[256];
  __shared__ float rm[B_SZ * NSEED];
  const int tid = threadIdx.x;
  const int R = B_SZ * NSEED;                  // 640 rows

  float part = 0.f;
  for (int i = tid; i < R; i += 256) {
    float s11 = 0.f, s10 = 0.f;
    for (int k = 0; k < KNN; ++k) {
      const float v = topd[i * KNN + k];
      s11 += v;
      if (k >= 1) s10 += v;
    }
    rm[i] = s11;
    part += s10;
  }
  red[tid] = part; __syncthreads();
  for (int off = 128; off > 0; off >>= 1) {
    if (tid < off) red[tid] += red[tid + off];
    __syncthreads();
  }
  const float overall_mean = red[0] / (float)(R * (KNN - 1));
  __syncthreads();

  float part2 = 0.f;
  for (int i = tid; i < R; i += 256) {
    const float m = rm[i] / ((float)KNN * overall_mean);
    rm[i] = m;
    part2 += m;
  }
  red[tid] = part2; __syncthreads();
  for (int off = 128; off > 0; off >>= 1) {
    if (tid < off) red[tid] += red[tid + off];
    __syncthreads();
  }
  const float mean_rm = red[0] / (float)R;
  __syncthreads();

  float part3 = 0.f;
  for (int i = tid; i < R; i += 256) {
    const float d = rm[i] - mean_rm;
    part3 += d * d;
  }
  red[tid] = part3; __syncthreads();
  for (int off = 128; off > 0; off >>= 1) {
    if (tid < off) red[tid] += red[tid + off];
    __syncthreads();
  }
  if (tid == 0) out[0] = red[0] / (float)(R - 1);   // ddof=1
}

extern "C" void kernel_launch(void* const* d_in, const int* in_sizes, int n_in,
                              void* d_out, int out_size, void* d_ws, size_t ws_size,
                              hipStream_t stream) {
  (void)in_sizes; (void)n_in; (void)out_size; (void)ws_size;
  const float* pcs = (const float*)d_in[0];          // [32,16384,3] f32
  float* ws     = (float*)d_ws;
  float* seedsB = ws;                                        // B*32*4
  float* cand   = seedsB + (size_t)B_SZ * 32 * 4;            // B*CH*20*11
  float* topd   = cand + (size_t)B_SZ * CH * NSEED * KNN;    // B*20*11
  float* out    = (float*)d_out;

  fps_kernel<<<B_SZ, 256, N_PTS * sizeof(float), stream>>>(pcs, seedsB);

  const size_t lds2 = (size_t)(CHUNK * NSEED + 8 * NSEED * KNN) * sizeof(float);
  knn_partial<<<dim3(CH, B_SZ), 256, lds2, stream>>>(pcs, seedsB, cand);

  knn_merge<<<B_SZ * NSEED, 64, 0, stream>>>(cand, topd);

  finalize_kernel<<<1, 256, 0, stream>>>(topd, out);
}